// SoftCLDiceLoss_87814901334658
// MI455X (gfx1250) — compile-verified
//
#include <hip/hip_runtime.h>
#include <hip/hip_bf16.h>

#define AS1 __attribute__((address_space(1)))
#define AS3 __attribute__((address_space(3)))

// CDNA5 async global->LDS path (ASYNCcnt).
#if __has_builtin(__builtin_amdgcn_global_load_async_to_lds_b32)
#define USE_ASYNC 1
#else
#define USE_ASYNC 0
#endif

// CDNA5 Tensor Data Mover path (TENSORcnt).
#if __has_builtin(__builtin_amdgcn_tensor_load_to_lds) && \
    __has_builtin(__builtin_amdgcn_s_wait_tensorcnt)
#define USE_TDM 1
#else
#define USE_TDM 0
#endif

typedef unsigned int u32;
typedef u32 v4u __attribute__((ext_vector_type(4)));
typedef int  v8i __attribute__((ext_vector_type(8)));
typedef int  v4i __attribute__((ext_vector_type(4)));

namespace {
constexpr int IMG_H = 1024;
constexpr int IMG_W = 1024;
constexpr int NIMG  = 16;              // B*C = 8*2
constexpr int SKEL_ITERS = 10;
constexpr int TILE  = 64;              // central output tile
constexpr int HALO  = 12;              // pool-chain depth = ITERS + 2
constexpr int R     = TILE + 2 * HALO; // 88
constexpr int RR    = R * R;           // 7744 cells per LDS buffer
constexpr int NTHREADS = 256;          // 8 wave32
constexpr float EPSF = 1e-6f;
}

// Async global->LDS copy of one float (tracked by ASYNCcnt on gfx1250).
__device__ __forceinline__ void async_load_b32(const float* g, float* l) {
#if USE_ASYNC
  __builtin_amdgcn_global_load_async_to_lds_b32((AS1 int*)g, (AS3 int*)l, 0, 0);
#else
  *l = *g;
#endif
}

__device__ __forceinline__ void wait_async_zero() {
#if USE_ASYNC
#if __has_builtin(__builtin_amdgcn_s_wait_asynccnt)
  __builtin_amdgcn_s_wait_asynccnt(0);
#else
  asm volatile("s_wait_asynccnt 0" ::: "memory");
#endif
#endif
}

#if USE_TDM
// One-shot DMA of an 88x88 fp32 tile (row stride IMG_W) from global into LDS.
// D# bitfields per CDNA5 ISA ch.8 (group0: count/lds/global/type,
// group1: data_size, tensor dims (set huge: interior tiles never go OOB),
// tile dims 88x88, dim0 stride 1024). Issued by one wave; EXEC is ignored.
__device__ __forceinline__ void tdm_load_tile(const float* g, float* lds) {
  const unsigned long long ga = (unsigned long long)(const void*)g;
  const u32 laddr = (u32)(size_t)(AS3 float*)lds;
  v4u g0;
  g0.x = 1u;                                   // count=1, user descriptor
  g0.y = laddr;                                // lds_addr (bytes)
  g0.z = (u32)(ga & 0xffffffffu);              // global_addr[31:0]
  g0.w = (u32)((ga >> 32) & 0x01ffffffu)       // global_addr[56:32]
       | (2u << 30);                           // type=2 ("image")
  const u32 TD = 1u << 20;                     // tensor_dim0/1: huge => no OOB
  v8i g1;
  g1[0] = 0x00020000;                          // data_size=2 (4 bytes)
  g1[1] = (int)((TD & 0xffffu) << 16);         // tensor_dim0[15:0] @ bits63:48
  g1[2] = (int)((TD >> 16) & 0xffffu);         // tensor_dim0[31:16] @ bits79:64
                                               // tensor_dim1[15:0] @ bits95:80
                                               //   = TD&0xffff = 0 (nothing to OR)
  g1[3] = (int)(((TD >> 16) & 0xffffu)         // tensor_dim1[31:16] @ bits111:96
       | ((u32)R << 16));                      // tile_dim0=88 @ bits127:112
  g1[4] = (int)R;                              // tile_dim1=88 (tile_dim2=0)
  g1[5] = IMG_W;                               // tensor_dim0_stride[31:0] = 1024
  g1[6] = 0;                                   // stride hi / dim1_stride lo
  g1[7] = 0;
  v4i gz4 = {0, 0, 0, 0};                      // groups 2/3 unused (2D tensor)
  v8i gz8 = {0, 0, 0, 0, 0, 0, 0, 0};
  // 6-arg form (this toolchain): (v4u, v8i, v4i, v4i, v8i, i32 cpol)
  __builtin_amdgcn_tensor_load_to_lds(g0, g1, gz4, gz4, gz8, 0);
}
#endif

template <bool ISMAX>
__device__ __forceinline__ float ext2(float a, float b) {
  return ISMAX ? fmaxf(a, b) : fminf(a, b);
}

// 3x3 max/min pool over the LDS region, clamp-to-edge at TRUE image borders
// (== -inf/+inf padded pooling), margin shrink at interior tile cuts.
// src valid at margin (m-1) -> dst valid at margin m.
// XSAFE: tile interior in x => branch-free inner loop (3 LDS reads, 1 write).
template <bool ISMAX, bool XSAFE>
__device__ __forceinline__ void pool3(const float* __restrict__ s,
                                      float* __restrict__ d,
                                      int m, int gy0, int gx0) {
  __syncthreads();                       // order vs previous producer
  const int lo = m, hi = R - m;
  const int n  = hi - lo;                // >= 64 always
  const int sub   = threadIdx.x & 3;     // 4 threads cooperate on a row
  const int chunk = (n + 3) >> 2;
  int c0 = lo + sub * chunk;
  int c1 = c0 + chunk; if (c1 > hi) c1 = hi;
  for (int row = lo + (int)(threadIdx.x >> 2); row < hi; row += NTHREADS / 4) {
    const int gy = gy0 + row;
    const float* r1 = s + row * R;
    const float* r0 = (gy > 0)         ? r1 - R : r1;   // clamp at image top
    const float* r2 = (gy < IMG_H - 1) ? r1 + R : r1;   // clamp at image bottom
    float* dr = d + row * R;
    if (c0 >= c1) continue;
    const int cm = XSAFE ? c0 - 1 : ((gx0 + c0 > 0) ? c0 - 1 : c0);
    float a = ext2<ISMAX>(ext2<ISMAX>(r0[cm], r1[cm]), r2[cm]);
    float b = ext2<ISMAX>(ext2<ISMAX>(r0[c0], r1[c0]), r2[c0]);
#pragma unroll 4
    for (int c = c0; c < c1; ++c) {
      const int cp = XSAFE ? c + 1 : ((gx0 + c < IMG_W - 1) ? c + 1 : c);
      float t = ext2<ISMAX>(ext2<ISMAX>(r0[cp], r1[cp]), r2[cp]);
      dr[c] = ext2<ISMAX>(ext2<ISMAX>(a, b), t);        // v_max3 / v_min3
      a = b; b = t;
    }
  }
}

template <bool ISMAX>
__device__ __forceinline__ void pool3d(bool xsafe, const float* s, float* d,
                                       int m, int gy0, int gx0) {
  if (xsafe) pool3<ISMAX, true >(s, d, m, gy0, gx0);
  else       pool3<ISMAX, false>(s, d, m, gy0, gx0);
}

// grid = (16, 16, 32): z = pass*16 + img.  pass 0: skeletonize p=sigmoid(pred),
// accumulate sum(skel_p*t), sum(skel_p).  pass 1: skeletonize t=target,
// accumulate sum(skel_t*p), sum(skel_t).   acc layout: [img][4].
__global__ __launch_bounds__(NTHREADS)
void clDiceSkelKernel(const float* __restrict__ pred,
                      const float* __restrict__ target,
                      float* __restrict__ acc) {
  __shared__ float smem[3 * RR];   // ~91 KB of the WGP's 320 KB LDS
  float* bufA = smem;
  float* bufB = smem + RR;
  float* bufC = smem + 2 * RR;

  const int tid  = threadIdx.x;
  const int img  = blockIdx.z & (NIMG - 1);
  const int pass = blockIdx.z >> 4;
  const int gx0  = blockIdx.x * TILE - HALO;
  const int gy0  = blockIdx.y * TILE - HALO;
  const size_t base = (size_t)img * ((size_t)IMG_H * IMG_W);

  const float* __restrict__ srcMain  = pass ? target : pred;
  const float* __restrict__ srcOther = pass ? pred   : target;

  const bool xsafe = (blockIdx.x >= 1) && (blockIdx.x <= 14);
  const bool ysafe = (blockIdx.y >= 1) && (blockIdx.y <= 14);

  // ---- Stage 1: halo'd region load ----
  bool tdmDone = false;
#if USE_TDM
  if (xsafe && ysafe) {            // fully interior: whole region in-image
    if (tid < 32) {                // one wave issues the tensor DMA
      tdm_load_tile(srcMain + base + (size_t)gy0 * IMG_W + gx0, bufA);
      __builtin_amdgcn_s_wait_tensorcnt(0);
    }
    tdmDone = true;
  }
#endif
  if (!tdmDone) {                  // edge tiles: clamped per-cell async copies
    for (int idx = tid; idx < RR; idx += NTHREADS) {
      int ry = idx / R;
      int rx = idx - ry * R;
      int gy = gy0 + ry; gy = (gy < 0) ? 0 : ((gy > IMG_H - 1) ? IMG_H - 1 : gy);
      int gx = gx0 + rx; gx = (gx < 0) ? 0 : ((gx > IMG_W - 1) ? IMG_W - 1 : gx);
      async_load_b32(srcMain + base + (size_t)gy * IMG_W + gx, &bufA[idx]);
    }
    wait_async_zero();
  }
  __syncthreads();

  // ---- Stage 2: pointwise transform (sigmoid for pred pass; clip both) ----
  for (int idx = tid; idx < RR; idx += NTHREADS) {
    float v = bufA[idx];
    if (pass == 0) v = 1.0f / (1.0f + __expf(-v));
    v = fminf(fmaxf(v, 0.0f), 1.0f);
    bufA[idx] = v;
  }
  // (pool3 entry barrier orders these writes)

  float* cur = bufA;   // img, valid margin grows with erosions
  float* t1  = bufB;
  float* t2  = bufC;

  // Each thread owns 16 central pixels: column px, rows py0 + 4*j.
  const int px  = tid & (TILE - 1);
  const int py0 = tid >> 6;            // 0..3
  float sk[16];

  // skel = relu(img - open(img))
  pool3d<false>(xsafe, cur, t1, 1, gy0, gx0);  // erode          (margin 1)
  pool3d<true >(xsafe, t1, t2, 2, gy0, gx0);   // dilate -> open (margin 2)
  __syncthreads();
#pragma unroll
  for (int j = 0; j < 16; ++j) {
    const int ci = (HALO + py0 + 4 * j) * R + (HALO + px);
    sk[j] = fmaxf(cur[ci] - t2[ci], 0.0f);
  }

  for (int k = 1; k <= SKEL_ITERS; ++k) {
    pool3d<false>(xsafe, cur, t1, k, gy0, gx0);     // img=erode(img) (margin k)
    { float* tmp = cur; cur = t1; t1 = tmp; }
    pool3d<false>(xsafe, cur, t1, k + 1, gy0, gx0); // erode(img)     (margin k+1)
    pool3d<true >(xsafe, t1, t2, k + 2, gy0, gx0);  // open(img)      (margin k+2)
    __syncthreads();
#pragma unroll
    for (int j = 0; j < 16; ++j) {
      const int ci = (HALO + py0 + 4 * j) * R + (HALO + px);
      float dlt = fmaxf(cur[ci] - t2[ci], 0.0f);
      sk[j] += fmaxf(dlt - sk[j] * dlt, 0.0f);
    }
  }

  // ---- Stage 3: per-tile partial sums against the partner field ----
  float ssum = 0.0f, swsum = 0.0f;
#pragma unroll
  for (int j = 0; j < 16; ++j) {
    const int gy = gy0 + HALO + py0 + 4 * j;   // always in-image (TILE | 1024)
    float o = srcOther[base + (size_t)gy * IMG_W + (gx0 + HALO + px)];
    if (pass == 1) o = 1.0f / (1.0f + __expf(-o));
    ssum  += sk[j];
    swsum += sk[j] * o;
  }
  // wave32 reduce, then block reduce, then 2 float atomics per block
#pragma unroll
  for (int off = 16; off > 0; off >>= 1) {
    ssum  += __shfl_xor(ssum,  off, 32);
    swsum += __shfl_xor(swsum, off, 32);
  }
  __syncthreads();                 // LDS reads above are done; reuse smem
  const int wid = tid >> 5;
  if ((tid & 31) == 0) { smem[wid] = ssum; smem[8 + wid] = swsum; }
  __syncthreads();
  if (tid == 0) {
    float s = 0.0f, sw = 0.0f;
#pragma unroll
    for (int w = 0; w < NTHREADS / 32; ++w) { s += smem[w]; sw += smem[8 + w]; }
    atomicAdd(&acc[img * 4 + pass * 2 + 0], sw);  // sum(skel * other)
    atomicAdd(&acc[img * 4 + pass * 2 + 1], s);   // sum(skel)
  }
}

__global__ void initAcc(float* acc) {
  acc[threadIdx.x] = 0.0f;         // 64 floats
}

__global__ void finalizeKernel(const float* __restrict__ acc,
                               float* __restrict__ out) {
  const int i = threadIdx.x;       // 32 threads, lanes 0..15 carry channels
  float cl = 0.0f;
  if (i < NIMG) {
    float tprec = acc[4 * i + 0] / (acc[4 * i + 1] + EPSF);
    float tsens = acc[4 * i + 2] / (acc[4 * i + 3] + EPSF);
    cl = 1.0f - 2.0f * tprec * tsens / (tprec + tsens + EPSF);
  }
#pragma unroll
  for (int off = 16; off > 0; off >>= 1) cl += __shfl_xor(cl, off, 32);
  if (i == 0) out[0] = cl * (1.0f / NIMG);
}

extern "C" void kernel_launch(void* const* d_in, const int* in_sizes, int n_in,
                              void* d_out, int out_size, void* d_ws, size_t ws_size,
                              hipStream_t stream) {
  (void)in_sizes; (void)n_in; (void)out_size; (void)ws_size;
  const float* pred   = (const float*)d_in[0];
  const float* target = (const float*)d_in[1];
  float* acc = (float*)d_ws;       // 64 floats of workspace
  float* out = (float*)d_out;

  hipLaunchKernelGGL(initAcc, dim3(1), dim3(64), 0, stream, acc);
  dim3 grid(IMG_W / TILE, IMG_H / TILE, NIMG * 2);
  hipLaunchKernelGGL(clDiceSkelKernel, grid, dim3(NTHREADS), 0, stream,
                     pred, target, acc);
  hipLaunchKernelGGL(finalizeKernel, dim3(1), dim3(32), 0, stream, acc, out);
}